// HeteroGNN_32530082300046
// MI455X (gfx1250) — compile-verified
//
#include <hip/hip_runtime.h>

// ---------------------------------------------------------------------------
// HeteroGNN (2-layer SAGE, mean aggr) for MI455X / gfx1250.
//   GEMMs   : V_WMMA_F32_16X16X4_F32, 16x16 tile/wave, A staged in LDS,
//             weights pre-packed so B fragment = one global_load_b64,
//             K/N compile-time -> all loads use immediate offsets.
//   Scatter : wave-per-edge, float4 gather + global_atomic_add_f32 (L2-resident)
// ---------------------------------------------------------------------------

typedef __attribute__((ext_vector_type(2))) float v2f;
typedef __attribute__((ext_vector_type(8))) float v8f;

#define NU 50000
#define NI 25000
#define HDIM 128

// ---------------------------------------------------------------------------
// Weight pack: Wp[(k>>1)*2N + col*2 + (k&1)] = W[k*N + col]
// so {W[k][col], W[k+1][col]} (k even) are adjacent -> single b64 load.
// ---------------------------------------------------------------------------
__global__ void __launch_bounds__(256)
pack_weights(const float* __restrict__ W, float* __restrict__ Wp, int K, int N)
{
    const int i = blockIdx.x * blockDim.x + threadIdx.x;
    if (i < K * N) {
        const int kk = i / N, col = i % N;
        Wp[((kk >> 1) * N + col) * 2 + (kk & 1)] = W[i];
    }
}

// ---------------------------------------------------------------------------
// K-chain of WMMAs for one 16x16 tile.
//   aLds : LDS ptr at [laneM*K + kHalf]   (per-k offset = k floats)
//   wp   : packed-W ptr at [col*2 + (lane>=16)*2N]  (per-k offset = k*N floats)
// All offsets compile-time immediates after full unroll.
// ---------------------------------------------------------------------------
template <int K, int N>
__device__ __forceinline__ void tile_mma(const float* __restrict__ aLds,
                                         const float* __restrict__ wp,
                                         v8f& acc)
{
#pragma unroll
    for (int k = 0; k < K; k += 4) {
        const v2f a = *(const v2f*)(aLds + k);
        const v2f b = *(const v2f*)(wp + (size_t)k * N);
        acc = __builtin_amdgcn_wmma_f32_16x16x4_f32(
            false, a, false, b, (short)0, acc, false, false);
    }
}

// ---------------------------------------------------------------------------
// C[M,N] = scale * (A1@W1 + (A2? A2@W2 : 0) + bias)  [+ C if ACC] [relu]
// Block = 256 thr = 8 waves. Block owns SPB=8/(N/16) strips of 16 rows;
// wave (strip, ntile) computes one 16x16 tile. A strips staged in LDS.
// Out-of-range rows are clamped (duplicate row M-1) -- they only affect
// output rows >= M which are never stored.
// ---------------------------------------------------------------------------
template <int K, int N>
__global__ void __launch_bounds__(256)
gemm_dual_wmma(const float* __restrict__ A1, const float* __restrict__ Wp1,
               const float* __restrict__ A2, const float* __restrict__ Wp2,
               const float* __restrict__ bias, float* __restrict__ C,
               int M, float scale, int accFlag, int reluFlag)
{
    constexpr int NT  = N / 16;     // N tiles (8 or 4)
    constexpr int SPB = 8 / NT;     // M strips per block (1 or 2)
    __shared__ __align__(16) float lds_a[SPB * 16 * K];

    const int lane   = threadIdx.x & 31;
    const int wave   = threadIdx.x >> 5;
    const int strip  = wave / NT;
    const int ntile  = wave % NT;
    const int mBase0 = blockIdx.x * (16 * SPB);

    const int laneM = lane & 15;
    const int kHalf = (lane >> 4) << 1;          // 0 or 2

    // ---- cooperative A staging (coalesced float4, rows clamped) ----------
    auto stage = [&](const float* __restrict__ A) {
        constexpr int nVec = SPB * 16 * (K / 4); // multiple of 256
#pragma unroll
        for (int i = threadIdx.x; i < nVec; i += 256) {
            const int r = i / (K / 4);
            const int c = i % (K / 4);
            int row = mBase0 + r;
            row = row < M ? row : (M - 1);
            const float4 v = *((const float4*)(A + (size_t)row * K) + c);
            *((float4*)(lds_a + r * K) + c) = v;
        }
    };

    const float* aFrag = lds_a + (strip * 16 + laneM) * K + kHalf;
    const int    col   = ntile * 16 + laneM;

    // bias pre-loaded into accumulator (D = A*B + C chains it through)
    const float bval = bias[col];
    v8f acc;
#pragma unroll
    for (int r = 0; r < 8; ++r) acc[r] = bval;

    stage(A1);
    __syncthreads();
    tile_mma<K, N>(aFrag, Wp1 + col * 2 + (lane >> 4) * 2 * N, acc);

    if (A2) {                                    // block-uniform branch
        __syncthreads();
        stage(A2);
        __syncthreads();
        tile_mma<K, N>(aFrag, Wp2 + col * 2 + (lane >> 4) * 2 * N, acc);
    }

    // ---- store: vgpr r -> row base+r (+8 for lanes 16..31), col ----------
    const int rowBase = mBase0 + strip * 16 + ((lane >> 4) << 3);
#pragma unroll
    for (int r = 0; r < 8; ++r) {
        const int rr = rowBase + r;
        if (rr < M) {
            const size_t idx = (size_t)rr * N + col;
            float v = acc[r] * scale;
            if (accFlag)  v += C[idx];
            if (reluFlag) v = fmaxf(v, 0.0f);
            C[idx] = v;
        }
    }
}

// ---------------------------------------------------------------------------
// One wave per edge: acc[dst] += X[src] (128 floats, float4 per lane),
// cnt[dst] += 1 (lane 0). acc/cnt must be pre-zeroed.
// ---------------------------------------------------------------------------
__global__ void __launch_bounds__(256)
scatter_accum(const float* __restrict__ X,
              const int* __restrict__ src, const int* __restrict__ dst,
              float* __restrict__ acc, float* __restrict__ cnt, int E)
{
    const int gwave = (int)((blockIdx.x * blockDim.x + threadIdx.x) >> 5);
    const int lane  = threadIdx.x & 31;
    if (gwave >= E) return;

    const int s = src[gwave];
    const int d = dst[gwave];
    const float4 v = *((const float4*)(X + (size_t)s * HDIM) + lane);
    float* ap = acc + (size_t)d * HDIM + lane * 4;
    atomicAdd(ap + 0, v.x);
    atomicAdd(ap + 1, v.y);
    atomicAdd(ap + 2, v.z);
    atomicAdd(ap + 3, v.w);
    if (lane == 0) atomicAdd(cnt + d, 1.0f);
}

// acc[i*128 + f] /= max(cnt[i], 1)
__global__ void __launch_bounds__(256)
normalize_mean(float* __restrict__ acc, const float* __restrict__ cnt, int n)
{
    const int total = n * HDIM;
    for (int i = blockIdx.x * blockDim.x + threadIdx.x; i < total;
         i += gridDim.x * blockDim.x) {
        const float c = cnt[i >> 7];
        acc[i] = acc[i] / fmaxf(c, 1.0f);
    }
}

// ---------------------------------------------------------------------------
// Host-side orchestration
// ---------------------------------------------------------------------------
template <int K, int N>
static inline void launch_gemm(const float* A1, const float* Wp1,
                               const float* A2, const float* Wp2,
                               const float* bias, float* C, int M, float scale,
                               int accF, int reluF, hipStream_t s)
{
    constexpr int SPB = 8 / (N / 16);
    const int blocks = (M + 16 * SPB - 1) / (16 * SPB);
    gemm_dual_wmma<K, N><<<blocks, 256, 0, s>>>(A1, Wp1, A2, Wp2, bias, C,
                                                M, scale, accF, reluF);
}

static inline void launch_pack(const float* W, float* Wp, int K, int N,
                               hipStream_t s)
{
    const int total = K * N;
    pack_weights<<<(total + 255) / 256, 256, 0, s>>>(W, Wp, K, N);
}

static inline void launch_scatter(const float* X, const int* src, const int* dst,
                                  float* acc, float* cnt, int E, hipStream_t s)
{
    const int blocks = (E * 32 + 255) / 256;
    scatter_accum<<<blocks, 256, 0, s>>>(X, src, dst, acc, cnt, E);
}

static inline void launch_norm(float* acc, const float* cnt, int n, hipStream_t s)
{
    const int blocks = (n * HDIM + 255) / 256;
    normalize_mean<<<blocks, 256, 0, s>>>(acc, cnt, n);
}

extern "C" void kernel_launch(void* const* d_in, const int* in_sizes, int n_in,
                              void* d_out, int out_size, void* d_ws, size_t ws_size,
                              hipStream_t stream)
{
    const float* x_user      = (const float*)d_in[0];
    const float* x_item      = (const float*)d_in[1];
    const int*   src_buys    = (const int*)d_in[2];
    const int*   dst_buys    = (const int*)d_in[3];
    const int*   src_bought  = (const int*)d_in[4];
    const int*   dst_bought  = (const int*)d_in[5];
    const int*   src_follows = (const int*)d_in[6];
    const int*   dst_follows = (const int*)d_in[7];
    const float* lin_u_W = (const float*)d_in[8];
    const float* lin_u_b = (const float*)d_in[9];
    const float* lin_i_W = (const float*)d_in[10];
    const float* lin_i_b = (const float*)d_in[11];
    const float* c1b_Wl  = (const float*)d_in[12];
    const float* c1b_bl  = (const float*)d_in[13];
    const float* c1b_Wr  = (const float*)d_in[14];
    const float* c1bb_Wl = (const float*)d_in[15];
    const float* c1bb_bl = (const float*)d_in[16];
    const float* c1bb_Wr = (const float*)d_in[17];
    const float* c1f_Wl  = (const float*)d_in[18];
    const float* c1f_bl  = (const float*)d_in[19];
    const float* c1f_Wr  = (const float*)d_in[20];
    const float* c2b_Wl  = (const float*)d_in[21];
    const float* c2b_bl  = (const float*)d_in[22];
    const float* c2b_Wr  = (const float*)d_in[23];
    const float* c2bb_Wl = (const float*)d_in[24];
    const float* c2bb_bl = (const float*)d_in[25];
    const float* c2bb_Wr = (const float*)d_in[26];
    const float* c2f_Wl  = (const float*)d_in[27];
    const float* c2f_bl  = (const float*)d_in[28];
    const float* c2f_Wr  = (const float*)d_in[29];

    const int EB  = in_sizes[2];   // 400000
    const int EBB = in_sizes[4];   // 400000
    const int EF  = in_sizes[6];   // 200000

    // ---- workspace carve-up (floats) --------------------------------------
    float* ws = (float*)d_ws;
    float* h_u    = ws;                          // [NU,128]
    float* h_i    = h_u    + (size_t)NU * HDIM;  // [NI,128]
    float* h_u2   = h_i    + (size_t)NI * HDIM;  // [NU,128]
    float* h_i2   = h_u2   + (size_t)NU * HDIM;  // [NI,128]
    float* acc_i  = h_i2   + (size_t)NI * HDIM;  // [NI,128]
    float* acc_u1 = acc_i  + (size_t)NI * HDIM;  // [NU,128]
    float* acc_u2 = acc_u1 + (size_t)NU * HDIM;  // [NU,128]
    float* cnt_i  = acc_u2 + (size_t)NU * HDIM;  // [NI]
    float* cnt_u1 = cnt_i  + NI;                 // [NU]
    float* cnt_u2 = cnt_u1 + NU;                 // [NU]
    // packed weights
    float* p = cnt_u2 + NU;
    float* wp_lin_u  = p; p += 128 * 128;
    float* wp_lin_i  = p; p += 64 * 128;
    float* wp_c1b_Wl = p; p += 128 * 128;
    float* wp_c1b_Wr = p; p += 128 * 128;
    float* wp_c1bb_Wl= p; p += 128 * 128;
    float* wp_c1bb_Wr= p; p += 128 * 128;
    float* wp_c1f_Wl = p; p += 128 * 128;
    float* wp_c1f_Wr = p; p += 128 * 128;
    float* wp_c2b_Wl = p; p += 128 * 64;
    float* wp_c2b_Wr = p; p += 128 * 64;
    float* wp_c2bb_Wl= p; p += 128 * 64;
    float* wp_c2bb_Wr= p; p += 128 * 64;
    float* wp_c2f_Wl = p; p += 128 * 64;
    float* wp_c2f_Wr = p; p += 128 * 64;

    float* out_u = (float*)d_out;                // [NU,64]
    float* out_i = out_u + (size_t)NU * 64;      // [NI,64]

    // ---- pack all weight matrices -----------------------------------------
    launch_pack(lin_u_W, wp_lin_u, 128, 128, stream);
    launch_pack(lin_i_W, wp_lin_i,  64, 128, stream);
    launch_pack(c1b_Wl,  wp_c1b_Wl,  128, 128, stream);
    launch_pack(c1b_Wr,  wp_c1b_Wr,  128, 128, stream);
    launch_pack(c1bb_Wl, wp_c1bb_Wl, 128, 128, stream);
    launch_pack(c1bb_Wr, wp_c1bb_Wr, 128, 128, stream);
    launch_pack(c1f_Wl,  wp_c1f_Wl,  128, 128, stream);
    launch_pack(c1f_Wr,  wp_c1f_Wr,  128, 128, stream);
    launch_pack(c2b_Wl,  wp_c2b_Wl,  128, 64, stream);
    launch_pack(c2b_Wr,  wp_c2b_Wr,  128, 64, stream);
    launch_pack(c2bb_Wl, wp_c2bb_Wl, 128, 64, stream);
    launch_pack(c2bb_Wr, wp_c2bb_Wr, 128, 64, stream);
    launch_pack(c2f_Wl,  wp_c2f_Wl,  128, 64, stream);
    launch_pack(c2f_Wr,  wp_c2f_Wr,  128, 64, stream);

    // ---- input projections -------------------------------------------------
    launch_gemm<128, 128>(x_user, wp_lin_u, nullptr, nullptr, lin_u_b, h_u,
                          NU, 1.0f, 0, 0, stream);
    launch_gemm<64, 128>(x_item, wp_lin_i, nullptr, nullptr, lin_i_b, h_i,
                         NI, 1.0f, 0, 0, stream);

    // ---- conv1 aggregation -------------------------------------------------
    hipMemsetAsync(acc_i,  0, (size_t)NI * HDIM * sizeof(float), stream);
    hipMemsetAsync(acc_u1, 0, (size_t)NU * HDIM * sizeof(float), stream);
    hipMemsetAsync(acc_u2, 0, (size_t)NU * HDIM * sizeof(float), stream);
    hipMemsetAsync(cnt_i,  0, (size_t)NI * sizeof(float), stream);
    hipMemsetAsync(cnt_u1, 0, (size_t)NU * sizeof(float), stream);
    hipMemsetAsync(cnt_u2, 0, (size_t)NU * sizeof(float), stream);

    launch_scatter(h_u, src_buys,    dst_buys,    acc_i,  cnt_i,  EB,  stream);
    launch_scatter(h_i, src_bought,  dst_bought,  acc_u1, cnt_u1, EBB, stream);
    launch_scatter(h_u, src_follows, dst_follows, acc_u2, cnt_u2, EF,  stream);

    launch_norm(acc_i,  cnt_i,  NI, stream);
    launch_norm(acc_u1, cnt_u1, NU, stream);
    launch_norm(acc_u2, cnt_u2, NU, stream);

    // ---- conv1 linear + relu ----------------------------------------------
    launch_gemm<128, 128>(acc_i, wp_c1b_Wl, h_i, wp_c1b_Wr, c1b_bl, h_i2,
                          NI, 1.0f, 0, 1, stream);
    launch_gemm<128, 128>(acc_u1, wp_c1bb_Wl, h_u, wp_c1bb_Wr, c1bb_bl, h_u2,
                          NU, 0.5f, 0, 0, stream);
    launch_gemm<128, 128>(acc_u2, wp_c1f_Wl, h_u, wp_c1f_Wr, c1f_bl, h_u2,
                          NU, 0.5f, 1, 1, stream);

    // ---- conv2 aggregation -------------------------------------------------
    hipMemsetAsync(acc_i,  0, (size_t)NI * HDIM * sizeof(float), stream);
    hipMemsetAsync(acc_u1, 0, (size_t)NU * HDIM * sizeof(float), stream);
    hipMemsetAsync(acc_u2, 0, (size_t)NU * HDIM * sizeof(float), stream);
    hipMemsetAsync(cnt_i,  0, (size_t)NI * sizeof(float), stream);
    hipMemsetAsync(cnt_u1, 0, (size_t)NU * sizeof(float), stream);
    hipMemsetAsync(cnt_u2, 0, (size_t)NU * sizeof(float), stream);

    launch_scatter(h_u2, src_buys,    dst_buys,    acc_i,  cnt_i,  EB,  stream);
    launch_scatter(h_i2, src_bought,  dst_bought,  acc_u1, cnt_u1, EBB, stream);
    launch_scatter(h_u2, src_follows, dst_follows, acc_u2, cnt_u2, EF,  stream);

    launch_norm(acc_i,  cnt_i,  NI, stream);
    launch_norm(acc_u1, cnt_u1, NU, stream);
    launch_norm(acc_u2, cnt_u2, NU, stream);

    // ---- conv2 linear (outputs, N=64) -------------------------------------
    launch_gemm<128, 64>(acc_i, wp_c2b_Wl, h_i2, wp_c2b_Wr, c2b_bl, out_i,
                         NI, 1.0f, 0, 0, stream);
    launch_gemm<128, 64>(acc_u1, wp_c2bb_Wl, h_u2, wp_c2bb_Wr, c2bb_bl, out_u,
                         NU, 0.5f, 0, 0, stream);
    launch_gemm<128, 64>(acc_u2, wp_c2f_Wl, h_u2, wp_c2f_Wr, c2f_bl, out_u,
                         NU, 0.5f, 1, 0, stream);
}